// MergeDistributions_68461778698487
// MI455X (gfx1250) — compile-verified
//
#include <hip/hip_runtime.h>

// Problem shape (fixed by the reference)
#define B_ 4
#define T_ 512
#define S_ 256
#define V_ 50257

// 64 KB + 16 B LDS accumulator tile per block -> up to 4 blocks resident per 320 KB WGP
#define CHUNK 16384
#define NCHUNK ((V_ + CHUNK - 1) / CHUNK)   // = 4 (last chunk covers 1105 cols)

// out[bt, v] = sum_{s : src[b,s]==v} p[bt, s]
// One block per (row bt, vocab chunk c). Phases:
//   1) zero LDS tile (ds_store_b128)
//   2) ds_add_f32 scatter of this row's 256 probabilities into the tile
//   3) async LDS->HBM stream: b128 bulk + b32 head/tail (ASYNCcnt path)
// V is odd, so row starts are only 4B-aligned; the LDS tile is shifted by
// (bt & 3) elements so global-16B-aligned addresses pair with LDS-16B-aligned
// addresses, enabling global_store_async_from_lds_b128 for the bulk.
__global__ void __launch_bounds__(256)
merge_distributions_scatter(const float* __restrict__ p,    // [B,T,S]
                            const int*   __restrict__ src,  // [B,S]
                            float* __restrict__ out)        // [B,T,V]
{
    __shared__ __align__(16) float lds[CHUNK + 4];

    const int tid   = threadIdx.x;
    const int bc    = blockIdx.x;
    const int c     = bc % NCHUNK;
    const int bt    = bc / NCHUNK;          // 0 .. B*T-1
    const int b     = bt / T_;
    const int lo    = c * CHUNK;
    const int n     = min(CHUNK, V_ - lo);  // valid columns in this chunk
    const int shift = bt & 3;               // LDS alignment shift (elements)
    const int head  = (4 - shift) & 3;      // leading unaligned globals

    // ---- Phase 1: zero the LDS accumulator tile (b128 DS stores) ----
    float4* lds4 = (float4*)lds;
    #pragma unroll 4
    for (int i = tid; i < (CHUNK + 4) / 4; i += 256)
        lds4[i] = make_float4(0.f, 0.f, 0.f, 0.f);
    __syncthreads();

    // ---- Phase 2: scatter-add this row's probs that land in [lo, lo+n) ----
    {
        const int s   = tid;                    // blockDim.x == S_ == 256
        const int idx = src[b * S_ + s];
        const int rel = idx - lo;
        if (rel >= 0 && rel < n) {
            const float val = p[(size_t)bt * S_ + s];
            atomicAdd(&lds[shift + rel], val);  // ds_add_f32 (no return)
        }
    }
    __syncthreads();                            // waits DScnt: tile is final

    // ---- Phase 3: stream finished tile LDS -> HBM via the async engine ----
    float* out_chunk = out + (size_t)bt * V_ + lo;   // elem offset ≡ bt (mod 4)

    // head: <=3 leading elements until the global address is 16B-aligned
    if (tid < head && tid < n) {
        float*   gp   = out_chunk + tid;
        unsigned loff = (unsigned)(unsigned long long)(const void*)&lds[shift + tid];
        asm volatile("global_store_async_from_lds_b32 %0, %1, off"
                     :: "v"(gp), "v"(loff) : "memory");
    }

    // bulk: 16B-aligned on both the global and (shifted) LDS side
    const int nmid = (n - head) >> 2;           // float4 groups
    for (int i = tid; i < nmid; i += 256) {
        const int v    = head + 4 * i;
        float*   gp    = out_chunk + v;
        unsigned loff  = (unsigned)(unsigned long long)(const void*)&lds[shift + v];
        asm volatile("global_store_async_from_lds_b128 %0, %1, off"
                     :: "v"(gp), "v"(loff) : "memory");
    }

    // tail: <=3 trailing elements
    const int done = head + 4 * nmid;
    const int rem  = n - done;
    if (tid < rem) {
        const int v    = done + tid;
        float*   gp    = out_chunk + v;
        unsigned loff  = (unsigned)(unsigned long long)(const void*)&lds[shift + v];
        asm volatile("global_store_async_from_lds_b32 %0, %1, off"
                     :: "v"(gp), "v"(loff) : "memory");
    }

    // Drain outstanding async LDS->global transfers before LDS is released.
    asm volatile("s_wait_asynccnt 0x0" ::: "memory");
}

// ---------------------------------------------------------------------------
// TDM syntax probe (compiled, never launched): tensor_store_from_lds with a
// hand-built 1-D D# (group0 + group1, VADDR2/3 omitted). If this assembles,
// the TDM engine becomes an option for the streaming phase in a later round.
// ---------------------------------------------------------------------------
typedef unsigned int u32x4 __attribute__((ext_vector_type(4)));
typedef unsigned int u32x8 __attribute__((ext_vector_type(8)));

__global__ void __launch_bounds__(32)
tdm_store_probe(float* out, int n)
{
    __shared__ __align__(16) float tile[CHUNK];

    unsigned long long ga  = (unsigned long long)(const void*)out;
    unsigned           la  = (unsigned)(unsigned long long)(const void*)&tile[0];
    unsigned           un  = (unsigned)n;

    u32x4 g0;
    g0.x = 1u;                                   // count=1, user descriptor
    g0.y = la;                                   // lds_addr (bytes)
    g0.z = (unsigned)ga;                         // global_addr[31:0]
    g0.w = ((unsigned)(ga >> 32) & 0x1FFFFFFu)   // global_addr[56:32]
         | (2u << 30);                           // type=2 ("image")

    u32x8 g1;
    g1.s0 = (2u << 16);                          // data_size=4B; no mask/flags
    g1.s1 = (un & 0xFFFFu) << 16;                // tensor_dim0[15:0]
    g1.s2 = (un >> 16) | (1u << 16);             // tensor_dim0[31:16], tensor_dim1=1
    g1.s3 = (un & 0xFFFFu) << 16;                // tensor_dim1 hi=0, tile_dim0=n
    g1.s4 = 0u;                                  // tile_dim1=0, tile_dim2=0
    g1.s5 = un;                                  // tensor_dim0_stride[31:0]
    g1.s6 = 0u;                                  // stride hi, tensor_dim1_stride lo
    g1.s7 = 0u;                                  // tensor_dim1_stride hi

    asm volatile("tensor_store_from_lds %0, %1" :: "s"(g0), "s"(g1) : "memory");
    asm volatile("s_wait_tensorcnt 0x0" ::: "memory");
}

extern "C" void kernel_launch(void* const* d_in, const int* in_sizes, int n_in,
                              void* d_out, int out_size, void* d_ws, size_t ws_size,
                              hipStream_t stream) {
    (void)in_sizes; (void)n_in; (void)out_size; (void)d_ws; (void)ws_size;

    const float* p   = (const float*)d_in[0];  // p_source_position [B,T,S] f32
    // d_in[1] (p_target_vocab) only defines V; unused by the math.
    const int*   src = (const int*)d_in[2];    // input_source [B,S] (int)
    float*       out = (float*)d_out;          // [B,T,V] f32

    dim3 grid(B_ * T_ * NCHUNK);               // 8192 blocks
    dim3 block(256);                           // 8 wave32 per block
    merge_distributions_scatter<<<grid, block, 0, stream>>>(p, src, out);
}